// DeconvSlimCapsule3D_59193239273838
// MI455X (gfx1250) — compile-verified
//
#include <hip/hip_runtime.h>
#include <math.h>

typedef float v2f __attribute__((ext_vector_type(2)));
typedef float v4f __attribute__((ext_vector_type(4)));
typedef float v8f __attribute__((ext_vector_type(8)));

#define NGROUP   16     // B * IN_DIM
#define IN_DIM   8
#define OCH      128    // OUT_DIM * OUT_ATOMS
#define DIN      16
#define NVOX     32768  // 32^3 output voxels

// ---------------------------------------------------------------------------
// Kernel 1: ConvTranspose3d(16->128, k4 s2 p1) as WMMA F32 16x16x4 GEMMs.
// grid = NGROUP * 8 parities * 16 d'  = 2048 blocks, 256 threads (8 waves).
// Wave i owns oc-tile [16*i, 16*i+16); A-weights live in VGPRs across 16 rows.
// LDS stores x with the (ic, ic+1) K-pair interleaved innermost so each B
// operand is ONE aligned ds_load_b64 straight into an even VGPR pair
// (no v_mov gathers in front of the WMMAs).
// Votes written as [b][voxel][in][oc] (voxel-major, 4KB contiguous per voxel).
// ---------------------------------------------------------------------------
__global__ __launch_bounds__(256)
void deconv_votes_wmma(const float* __restrict__ x,
                       const float* __restrict__ w,
                       const float* __restrict__ deconv_b,
                       float* __restrict__ votes) {
  // [cpair 8][sd 2][sh 2][col 18][pair 2]  (cpair = ic>>1), 1152 floats
  __shared__ float xs[8 * 2 * 2 * 18 * 2];

  const int tid   = threadIdx.x;
  const int lane  = tid & 31;
  const int wv    = tid >> 5;        // wave id == oc tile id (0..7)
  const int ocb   = wv * 16;
  const int m     = lane & 15;       // A row (oc within tile) / B,N column
  const int khalf = lane >> 4;       // K half: lanes 0-15 -> K0,K1 ; 16-31 -> K2,K3

  const int wg  = blockIdx.x;
  const int g   = wg >> 7;           // 0..15 = b*8 + in_dim
  const int rem = wg & 127;
  const int par = rem >> 4;          // parity class 0..7
  const int dp  = rem & 15;          // d'
  const int pd = (par >> 2) & 1, ph = (par >> 1) & 1, pw = par & 1;
  const int bidx = g >> 3, indim = g & 7;

  // Valid taps for this parity: kd = 2*s + (1-pd), input delta per s below.
  const int deltad[2] = { pd ? 1 : 0, pd ? 0 : -1 };
  const int deltah[2] = { ph ? 1 : 0, ph ? 0 : -1 };
  const int deltaw[2] = { pw ? 1 : 0, pw ? 0 : -1 };

  // --- load per-wave A operands: A[m][k] = w[ic_base+k][ocb+m][kd,kh,kw] ---
  v2f A[2][2][2][4];
  {
    const int kd0 = 1 - pd, kh0 = 1 - ph, kw0 = 1 - pw;
    for (int sd = 0; sd < 2; ++sd)
      for (int sh = 0; sh < 2; ++sh)
        for (int sw = 0; sw < 2; ++sw) {
          const int tap = ((2 * sd + kd0) * 4 + (2 * sh + kh0)) * 4 + (2 * sw + kw0);
          for (int c = 0; c < 4; ++c) {
            const int ic = c * 4 + khalf * 2;
            const float* wp = w + (size_t)(ic * OCH + ocb + m) * 64 + tap;
            A[sd][sh][sw][c].x = wp[0];
            A[sd][sh][sw][c].y = wp[OCH * 64];   // ic+1
          }
        }
  }
  float bl[8];
  for (int j = 0; j < 8; ++j) bl[j] = deconv_b[ocb + khalf * 8 + j];

  // zero halo columns (cols 0 and 17, both pair elements, all 32 rows) once
  if (tid < 128) {
    const int e = tid & 1, cs = (tid >> 1) & 1, r = tid >> 2;  // r = 0..31
    xs[(r * 18 + (cs ? 17 : 0)) * 2 + e] = 0.f;
  }

  const size_t xg = (size_t)g * 16 * DIN * DIN * DIN;

  for (int hp = 0; hp < 16; ++hp) {
    __syncthreads();
    // cooperative load: 64 rows x 16 cols; thread t -> row t>>2, 4 cols
    {
      const int row  = tid >> 2;
      const int colq = (tid & 3) * 4;
      const int ic = row >> 2, sd = (row >> 1) & 1, sh = row & 1;
      const int cp = ic >> 1, e = ic & 1;
      const int d = dp + deltad[sd];
      const int h = hp + deltah[sh];
      const bool ok = ((unsigned)d < 16u) && ((unsigned)h < 16u);
      const float* src = x + xg + (((size_t)ic * DIN + (ok ? d : 0)) * DIN + (ok ? h : 0)) * DIN + colq;
      float* dst = xs + (((cp * 4 + sd * 2 + sh) * 18) + 1 + colq) * 2 + e;
      for (int j = 0; j < 4; ++j) dst[j * 2] = ok ? src[j] : 0.f;
    }
    __syncthreads();

    v8f acc = {0.f, 0.f, 0.f, 0.f, 0.f, 0.f, 0.f, 0.f};
    const int n = lane & 15;                 // output position (w') within tile
    for (int sd = 0; sd < 2; ++sd)
      for (int sh = 0; sh < 2; ++sh)
        for (int sw = 0; sw < 2; ++sw) {
          const int col = 1 + n + deltaw[sw];
          for (int c = 0; c < 4; ++c) {
            const int cp = c * 2 + khalf;    // (ic, ic+1) pair, ic = 4c + 2*khalf
            const v2f bvec = *(const v2f*)&xs[((cp * 4 + sd * 2 + sh) * 18 + col) * 2];
            acc = __builtin_amdgcn_wmma_f32_16x16x4_f32(
                false, A[sd][sh][sw][c], false, bvec, (short)0, acc, false, false);
          }
        }

    for (int j = 0; j < 8; ++j) acc[j] += bl[j];

    const int od = 2 * dp + pd, oh = 2 * hp + ph, ow = 2 * n + pw;
    const int o  = (od * 32 + oh) * 32 + ow;
    float* dst = votes + ((size_t)(bidx * NVOX + o) * IN_DIM + indim) * OCH
                       + ocb + khalf * 8;
    ((v4f*)dst)[0] = *((const v4f*)&acc);
    ((v4f*)dst)[1] = *((const v4f*)&acc + 1);
  }
}

// ---------------------------------------------------------------------------
// Kernel 2: Sabour routing (3 iters) + squash.
// lane = out*4 + v ; 8 lanes per voxel, softmax over out via shfl_xor 4/8/16.
// grid = 65536 voxels / 32 per block = 2048 blocks, 256 threads.
// ---------------------------------------------------------------------------
__global__ __launch_bounds__(256)
void routing_squash(const float* __restrict__ votes,
                    const float* __restrict__ rbias,
                    float* __restrict__ out) {
  const int tid  = threadIdx.x;
  const int lane = tid & 31;
  const int wv   = tid >> 5;
  const int outd = lane >> 2;                      // 0..7
  const int v    = lane & 3;
  const int vox  = blockIdx.x * 32 + wv * 4 + v;   // 0..65535 == b*32768 + o
  const int b    = vox >> 15;
  const int o    = vox & 32767;

  float vt[8][16];
  const float* base = votes + (size_t)vox * IN_DIM * OCH + outd * 16;
  for (int in = 0; in < 8; ++in) {
    const v4f* p = (const v4f*)(base + (size_t)in * OCH);
    v4f q0 = p[0], q1 = p[1], q2 = p[2], q3 = p[3];
    for (int a = 0; a < 4; ++a) {
      vt[in][a] = q0[a]; vt[in][4 + a] = q1[a];
      vt[in][8 + a] = q2[a]; vt[in][12 + a] = q3[a];
    }
  }
  float rb[16];
  for (int a = 0; a < 16; ++a) rb[a] = rbias[outd * 16 + a];

  float n2[8];
  for (int in = 0; in < 8; ++in) {
    float s = 0.f;
    for (int a = 0; a < 16; ++a) s += vt[in][a] * vt[in][a];
    n2[in] = sqrtf(s);
  }

  float logits[8];
  for (int in = 0; in < 8; ++in) logits[in] = 0.f;
  float pa[16];

  for (int it = 0; it < 3; ++it) {
    float route[8];
    for (int in = 0; in < 8; ++in) {            // softmax over out capsules
      float e = __expf(logits[in]);
      float t = e;
      t += __shfl_xor(t, 4, 32);
      t += __shfl_xor(t, 8, 32);
      t += __shfl_xor(t, 16, 32);
      route[in] = e / t;
    }
    for (int a = 0; a < 16; ++a) pa[a] = rb[a];
    for (int in = 0; in < 8; ++in)
      for (int a = 0; a < 16; ++a) pa[a] += vt[in][a] * route[in];
    if (it < 2) {
      float s = 0.f;
      for (int a = 0; a < 16; ++a) s += pa[a] * pa[a];
      const float n1 = sqrtf(s);
      for (int in = 0; in < 8; ++in) {
        float dot = 0.f;
        for (int a = 0; a < 16; ++a) dot += pa[a] * vt[in][a];
        logits[in] += dot / fmaxf(n1 * n2[in], 1e-8f);
      }
    }
  }

  float nsq = 0.f;
  for (int a = 0; a < 16; ++a) nsq += pa[a] * pa[a];
  const float norm  = sqrtf(nsq);
  const float scale = (1.f / (norm + 1e-12f)) * (nsq / (1.f + nsq));
  for (int a = 0; a < 16; ++a)
    out[(((size_t)(b * 8 + outd) * 16 + a) << 15) + o] = pa[a] * scale;
}

extern "C" void kernel_launch(void* const* d_in, const int* in_sizes, int n_in,
                              void* d_out, int out_size, void* d_ws, size_t ws_size,
                              hipStream_t stream) {
  (void)in_sizes; (void)n_in; (void)out_size; (void)ws_size;
  const float* x  = (const float*)d_in[0];
  const float* w  = (const float*)d_in[1];
  const float* db = (const float*)d_in[2];
  const float* rb = (const float*)d_in[3];
  float* votes = (float*)d_ws;          // 2*8*8*16*32^3 fp32 = 256 MiB
  float* outp  = (float*)d_out;

  deconv_votes_wmma<<<NGROUP * 8 * 16, 256, 0, stream>>>(x, w, db, votes);
  routing_squash<<<2048, 256, 0, stream>>>(votes, rb, outp);
}